// LowLevelRNN_classifier_6932077216197
// MI455X (gfx1250) — compile-verified
//
#include <hip/hip_runtime.h>
#include <hip/hip_bf16.h>
#include <math.h>

// ---------------------------------------------------------------------------
// LowLevelRNN classifier for MI455X (gfx1250, wave32, WMMA f32, async-to-LDS)
//   Phase 1: XW = gather(emb_table, seq) @ Wx + bh   (WMMA GEMM) -> d_out stage
//   Phase 2: h_t = tanh(XW[t] + h_{t-1} @ Wh) x 8192 (L2-resident MV)
//   Phase 3: Y  = H_all @ Wy + by                    (WMMA GEMM) -> d_out
// d_ws layout: [ H_all : T*H floats | h ping-pong : 2*H floats ]  (~67.1 MB)
// ---------------------------------------------------------------------------

typedef float v2f __attribute__((ext_vector_type(2)));
typedef float v8f __attribute__((ext_vector_type(8)));

#define USE_ASYNC_LDS 1

#define KC   16     // K chunk per LDS panel
#define BM   128    // block tile M
#define BN   64     // block tile N
#define APAD 20     // A panel row stride (floats): 16B-aligned, bank-conflict-free

// Stage one float4 from global into LDS.
// Async path: GLOBAL_LOAD_ASYNC_TO_LDS_B128 (ASYNCcnt), lane-parallel DMA into
// LDS without touching VGPRs. LDS operand is the low 32 bits of the generic
// address (== LDS byte offset).
__device__ __forceinline__ void stage16(float* l, const float* g) {
#if USE_ASYNC_LDS
  asm volatile("global_load_async_to_lds_b128 %0, %1, off"
               :: "v"((unsigned)(uintptr_t)l), "v"(g)
               : "memory");
#else
  *(float4*)l = *(const float4*)g;
#endif
}
__device__ __forceinline__ void async_wait_le3() {
#if USE_ASYNC_LDS
  asm volatile("s_wait_asynccnt 0x3" ::: "memory");
#endif
}
__device__ __forceinline__ void async_wait_0() {
#if USE_ASYNC_LDS
  asm volatile("s_wait_asynccnt 0x0" ::: "memory");
#endif
}

__device__ __forceinline__ v8f wmma4(v2f a, v2f b, v8f c) {
  return __builtin_amdgcn_wmma_f32_16x16x4_f32(false, a, false, b,
                                               (short)0, c, false, false);
}

// C[M,N] = A[M,K] @ B[K,N] + bias[N]; if idx != nullptr, row m of A is
// table[idx[m]]. 256 threads = 8 waves in a 4(M)x2(N) wave grid; each wave owns
// a 32x32 output tile = 4 f32 16x16 accumulators (4 WMMAs per K-step of 4,
// 2 A-frags x 2 B-frags reuse). Double-buffered LDS panels fed by async DMA.
__global__ __launch_bounds__(256) void gemm128x64_wmma_f32(
    const float* __restrict__ A, const float* __restrict__ Bmat,
    const float* __restrict__ bias, float* __restrict__ C,
    int M, int N, int K,
    const int* __restrict__ idx, const float* __restrict__ table)
{
  __shared__ float As[2][BM * APAD];   // 2 x 10240 B
  __shared__ float Bs[2][KC * BN];     // 2 x 4096 B

  const int tid  = threadIdx.x;
  const int lane = tid & 31;
  const int wave = tid >> 5;
  const int wm   = (wave & 3) << 5;    // wave M offset: 0,32,64,96
  const int wn   = (wave >> 2) << 5;   // wave N offset: 0,32
  const int m0   = blockIdx.y * BM;
  const int n0   = blockIdx.x * BN;

  // A staging: 512 float4 per chunk -> thread covers f4 #tid and #tid+256
  const int a_row0 = tid >> 2;                 // 0..63
  const int a_row1 = a_row0 + 64;              // 64..127
  const int a_col  = (tid & 3) << 2;           // 0,4,8,12
  const float* a_src0;
  const float* a_src1;
  if (idx) {
    a_src0 = table + (size_t)idx[m0 + a_row0] * (size_t)K;
    a_src1 = table + (size_t)idx[m0 + a_row1] * (size_t)K;
  } else {
    a_src0 = A + (size_t)(m0 + a_row0) * (size_t)K;
    a_src1 = A + (size_t)(m0 + a_row1) * (size_t)K;
  }
  // B staging: 256 float4 per chunk -> thread covers f4 #tid
  const int b_row = tid >> 4;                  // 0..15
  const int b_col = (tid & 15) << 2;           // 0..60

  // WMMA fragment addressing (ISA 7.12.2, 32-bit operands, wave32):
  //   lanes 0-15: M/N = lane,    K pair {k, k+1}
  //   lanes16-31: M/N = lane-16, K pair {k+2, k+3}
  const int arA  = wm + (lane & 15);
  const int arB  = arA + 16;
  const int ksel = (lane >> 4) << 1;           // 0 or 2
  const int bnn  = wn + (lane & 15);

  v8f acc00 = {0.f,0.f,0.f,0.f,0.f,0.f,0.f,0.f};
  v8f acc01 = acc00, acc10 = acc00, acc11 = acc00;

  auto stage_chunk = [&](int buf, int k0) {
    stage16(&As[buf][a_row0 * APAD + a_col], a_src0 + k0 + a_col);
    stage16(&As[buf][a_row1 * APAD + a_col], a_src1 + k0 + a_col);
    stage16(&Bs[buf][b_row * BN + b_col],
            Bmat + (size_t)(k0 + b_row) * (size_t)N + (size_t)(n0 + b_col));
  };

  stage_chunk(0, 0);                           // prologue

  for (int k0 = 0; k0 < K; k0 += KC) {
    const int  buf      = (k0 / KC) & 1;
    const bool has_next = (k0 + KC) < K;
    if (has_next) stage_chunk(buf ^ 1, k0 + KC);   // overlap with compute wait
    if (has_next) async_wait_le3();                // cur chunk's 3 have landed
    else          async_wait_0();
    __syncthreads();                               // all waves' panel complete

    const float* Ab = &As[buf][0];
    const float* Bb = &Bs[buf][0];
#pragma unroll
    for (int kk = 0; kk < KC; kk += 4) {
      v2f a0, a1, b0, b1;
      a0.x = Ab[arA * APAD + kk + ksel];
      a0.y = Ab[arA * APAD + kk + ksel + 1];
      a1.x = Ab[arB * APAD + kk + ksel];
      a1.y = Ab[arB * APAD + kk + ksel + 1];
      b0.x = Bb[(kk + ksel) * BN + bnn];
      b0.y = Bb[(kk + ksel + 1) * BN + bnn];
      b1.x = Bb[(kk + ksel) * BN + bnn + 16];
      b1.y = Bb[(kk + ksel + 1) * BN + bnn + 16];
      acc00 = wmma4(a0, b0, acc00);
      acc01 = wmma4(a0, b1, acc01);
      acc10 = wmma4(a1, b0, acc10);
      acc11 = wmma4(a1, b1, acc11);
    }
    __syncthreads();                               // release buf for overwrite
  }

  // C/D layout: VGPR r -> row r (lanes 0-15) / row r+8 (lanes 16-31)
  const int mrow = m0 + wm + ((lane >> 4) << 3);
  const int ncol = n0 + wn + (lane & 15);
  const float bia0 = bias[ncol];
  const float bia1 = bias[ncol + 16];
#pragma unroll
  for (int r = 0; r < 8; ++r) {
    float* c0 = C + (size_t)(mrow + r) * (size_t)N + ncol;
    float* c1 = C + (size_t)(mrow + 16 + r) * (size_t)N + ncol;
    c0[0]  = acc00[r] + bia0;
    c0[16] = acc01[r] + bia1;
    c1[0]  = acc10[r] + bia0;
    c1[16] = acc11[r] + bia1;
  }
}

// One recurrence step: h_out[j] = tanh(xw[j] + sum_k h_in[k] * Wh[k*H + j]).
// 64 blocks x 256 threads: block owns 32 consecutive j columns, 8-way k-split,
// h_in staged in LDS, deterministic LDS tree reduction. Every Wh element is
// read exactly once per step, coalesced, L2-resident (Wh = 16.8 MB << 192 MB).
__global__ __launch_bounds__(256) void rnn_step(
    const float* __restrict__ Wh, const float* __restrict__ xw,
    const float* __restrict__ h_in, float* __restrict__ h_out,
    float* __restrict__ hall_row, float* __restrict__ hfinal)
{
  const int H = 2048;
  __shared__ float hs[2048];
  __shared__ float psum[256];

  const int tid = threadIdx.x;
  for (int i = tid; i < H / 4; i += 256)
    *(float4*)&hs[i * 4] = *(const float4*)&h_in[i * 4];
  __syncthreads();

  const int jl = tid & 31;         // column within block tile
  const int ks = tid >> 5;         // k-slice 0..7 (256 k each)
  const int j  = blockIdx.x * 32 + jl;

  const float* p = Wh + (size_t)(ks * 256) * (size_t)H + (size_t)j;
  float sum = 0.f;
#pragma unroll 8
  for (int k = 0; k < 256; ++k) {
    sum += hs[ks * 256 + k] * (*p);
    p += H;
  }
  psum[tid] = sum;
  __syncthreads();

  if (ks == 0) {
    float s = psum[jl];
#pragma unroll
    for (int r = 1; r < 8; ++r) s += psum[r * 32 + jl];
    const float h = tanhf(s + xw[j]);
    h_out[j]    = h;
    hall_row[j] = h;
    if (hfinal) hfinal[j] = h;
  }
}

__global__ __launch_bounds__(256) void copy_h0(const float* __restrict__ src,
                                               float* __restrict__ dst)
{
  const int i = blockIdx.x * 256 + threadIdx.x;
  dst[i] = src[i];
}

extern "C" void kernel_launch(void* const* d_in, const int* in_sizes, int n_in,
                              void* d_out, int out_size, void* d_ws, size_t ws_size,
                              hipStream_t stream) {
  (void)in_sizes; (void)n_in; (void)out_size; (void)ws_size;

  const int T = 8192, H = 2048, V = 2048, Ydim = 2048;

  const int*   seq   = (const int*)  d_in[0];  // (T,1) token ids
  const float* h0    = (const float*)d_in[1];  // (H,)
  const float* table = (const float*)d_in[2];  // (V,V) emb_table
  const float* Wx    = (const float*)d_in[3];  // (V,H)
  const float* Wh    = (const float*)d_in[4];  // (H,H)
  const float* Wy    = (const float*)d_in[5];  // (H,Y)
  const float* bh    = (const float*)d_in[6];  // (H,)
  const float* by    = (const float*)d_in[7];  // (Y,)

  float* out  = (float*)d_out;                 // ys (T*Y) then h_final (H)
  float* hall = (float*)d_ws;                  // T*H floats
  float* hbuf = hall + (size_t)T * H;          // 2*H floats, ping-pong

  // h_{-1} = hidden_state
  copy_h0<<<dim3(H / 256), dim3(256), 0, stream>>>(h0, hbuf);

  // Phase 1: XW = gather(emb) @ Wx + bh -> staged in d_out (consumed by phase 2
  // before phase 3 overwrites it)
  gemm128x64_wmma_f32<<<dim3(H / BN, T / BM), dim3(256), 0, stream>>>(
      nullptr, Wx, bh, out, T, H, V, seq, table);

  // Phase 2: sequential scan; stream ordering carries the h dependency
  for (int t = 0; t < T; ++t) {
    rnn_step<<<dim3(H / 32), dim3(256), 0, stream>>>(
        Wh,
        out  + (size_t)t * H,                 // xw_t
        hbuf + (size_t)(t & 1) * H,           // h_{t-1}
        hbuf + (size_t)((t + 1) & 1) * H,     // h_t
        hall + (size_t)t * H,                 // H_all[t]
        (t == T - 1) ? (out + (size_t)T * Ydim) : nullptr);  // h_final tail
  }

  // Phase 3: Y = H_all @ Wy + by -> d_out (overwrites the XW staging)
  gemm128x64_wmma_f32<<<dim3(Ydim / BN, T / BM), dim3(256), 0, stream>>>(
      hall, Wy, by, out, T, Ydim, H, nullptr, nullptr);
}